// DKNLoss_10153302688165
// MI455X (gfx1250) — compile-verified
//
#include <hip/hip_runtime.h>
#include <math.h>

typedef __attribute__((ext_vector_type(2))) float v2f;
typedef __attribute__((ext_vector_type(8))) float v8f;

#define D_LAT   128
#define K_CL    256
#define RS_PAD  132      // R row stride in LDS (floats): bank = (4k + d) % 64
#define ES_PAD  260      // e row stride in LDS (floats): bank = (4r + k) % 64
#define WAVES   8

// ---------------------------------------------------------------- zero output
__global__ void zero_kernel(float* out) { out[0] = 0.0f; }

// ------------------------------------------------- reconstruction MSE (HBM-bound)
__global__ __launch_bounds__(256) void recon_kernel(const float4* __restrict__ x,
                                                    const float4* __restrict__ ax,
                                                    float* __restrict__ out,
                                                    int n4, float scale) {
    float acc = 0.0f;
    int i = blockIdx.x * blockDim.x + threadIdx.x;
    const int stride = gridDim.x * blockDim.x;
    for (; i < n4; i += stride) {
        float4 a = x[i];
        float4 b = ax[i];
        float dx = a.x - b.x, dy = a.y - b.y, dz = a.z - b.z, dw = a.w - b.w;
        acc = fmaf(dx, dx, acc);
        acc = fmaf(dy, dy, acc);
        acc = fmaf(dz, dz, acc);
        acc = fmaf(dw, dw, acc);
    }
    // wave32 reduction
    for (int off = 16; off > 0; off >>= 1) acc += __shfl_down(acc, off, 32);
    __shared__ float red[8];
    const int w = threadIdx.x >> 5, ln = threadIdx.x & 31;
    if (ln == 0) red[w] = acc;
    __syncthreads();
    if (threadIdx.x == 0) {
        float s = 0.0f;
        for (int k = 0; k < 8; ++k) s += red[k];
        atomicAdd(out, s * scale);
    }
}

// ----------------------------------------------- clustering loss (WMMA f32 path)
// Per block: 8 waves, each wave owns a 16-row tile of h_x.
// GEMM1: dist2 = |h|^2 + |R|^2 - 2 h.R^T   (16x256, K=128, f32 WMMA 16x16x4)
// e = exp(-dist); GEMM2: W = e.R (16x128, K=256); weighted = W / (rowsum(e)+eps)
__global__ __launch_bounds__(256, 1) void cluster_kernel(const float* __restrict__ h,
                                                         const float* __restrict__ R,
                                                         float* __restrict__ out,
                                                         float scale) {
    __shared__ float sR[K_CL * RS_PAD];        // 135,168 B  R[k][d], padded
    __shared__ float sr2[K_CL];                //   1,024 B  |R_k|^2
    __shared__ float sE[WAVES][16 * ES_PAD];   // 133,120 B  e tiles, per wave
    __shared__ float sH2[WAVES][16];           //     512 B  |h_m|^2
    __shared__ float sRS[WAVES][16];           //     512 B  row sums of e
    __shared__ float sRed[WAVES];

    const int tid  = threadIdx.x;
    const int w    = tid >> 5;
    const int ln   = tid & 31;
    const int lnm  = ln & 15;       // position within half-wave
    const int half = ln >> 4;       // 0: K%4 in {0,1}; 1: K%4 in {2,3}

    // ---- stage R into LDS (coalesced, conflict-free)
    for (int idx = tid; idx < K_CL * D_LAT; idx += 256) {
        const int k = idx >> 7, d = idx & 127;
        sR[k * RS_PAD + d] = R[idx];
    }
    __syncthreads();

    // ---- |R_k|^2
    {
        float s = 0.0f;
        const float* rp = &sR[tid * RS_PAD];
        #pragma unroll 8
        for (int d = 0; d < D_LAT; ++d) s = fmaf(rp[d], rp[d], s);
        sr2[tid] = s;
    }
    __syncthreads();

    const int m0 = (blockIdx.x * WAVES + w) * 16;

    // ---- load A-operand tile of h into registers per the 16x4 f32 WMMA layout:
    // lanes 0-15 hold rows 0-15 with K%4 in {0,1}; lanes 16-31 rows 0-15, K%4 in {2,3}
    float a[64];
    {
        const float* hp = h + (size_t)(m0 + lnm) * D_LAT + 2 * half;
        float h2p = 0.0f;
        #pragma unroll
        for (int i = 0; i < 32; ++i) {
            const float2 t = *(const float2*)(hp + 4 * i);
            a[2 * i]     = t.x;
            a[2 * i + 1] = t.y;
            h2p = fmaf(t.x, t.x, h2p);
            h2p = fmaf(t.y, t.y, h2p);
        }
        const float h2 = h2p + __shfl_xor(h2p, 16, 32);   // combine the two halves
        if (half == 0) sH2[w][lnm] = h2;
    }

    // ---- GEMM1 over 16 cluster tiles: h (16x128) x R^T (128x16)
    float rsum[8] = {0.f, 0.f, 0.f, 0.f, 0.f, 0.f, 0.f, 0.f};
    for (int nt = 0; nt < 16; ++nt) {
        const int n = nt * 16 + lnm;
        v8f acc = {};
        const float* bp = &sR[n * RS_PAD + 2 * half];     // B[k][n] = R[n][k]
        #pragma unroll
        for (int kk = 0; kk < 32; ++kk) {
            v2f av; av.x = a[2 * kk]; av.y = a[2 * kk + 1];
            v2f bv; bv.x = bp[4 * kk]; bv.y = bp[4 * kk + 1];
            acc = __builtin_amdgcn_wmma_f32_16x16x4_f32(false, av, false, bv,
                                                        (short)0, acc, false, false);
        }
        const float r2n = sr2[n];
        #pragma unroll
        for (int j = 0; j < 8; ++j) {
            const int m = j + half * 8;                   // D-tile row for this lane
            float d2 = sH2[w][m] + r2n - 2.0f * acc[j];
            d2 = fmaxf(d2, 0.0f);
            const float e = expf(-sqrtf(d2));             // ALPHA = 1
            sE[w][m * ES_PAD + n] = e;
            rsum[j] += e;
        }
    }

    // ---- row sums of e: reduce across the 16 lanes of each half-wave
    #pragma unroll
    for (int j = 0; j < 8; ++j) {
        float v = rsum[j];
        v += __shfl_xor(v, 1, 32);
        v += __shfl_xor(v, 2, 32);
        v += __shfl_xor(v, 4, 32);
        v += __shfl_xor(v, 8, 32);
        if (lnm == 0) sRS[w][half * 8 + j] = v;
    }

    // ---- GEMM2: W = e (16x256) x R (256x16 tiles), then normalize + MSE
    float cl = 0.0f;
    const float* ap = &sE[w][lnm * ES_PAD + 2 * half];    // A-operand rows of e
    for (int dt = 0; dt < 8; ++dt) {
        const int d = dt * 16 + lnm;
        v8f acc = {};
        #pragma unroll 8
        for (int kk = 0; kk < 64; ++kk) {
            v2f av; av.x = ap[4 * kk]; av.y = ap[4 * kk + 1];
            const float* bp = &sR[(4 * kk + 2 * half) * RS_PAD + d];
            v2f bv; bv.x = bp[0]; bv.y = bp[RS_PAD];
            acc = __builtin_amdgcn_wmma_f32_16x16x4_f32(false, av, false, bv,
                                                        (short)0, acc, false, false);
        }
        #pragma unroll
        for (int j = 0; j < 8; ++j) {
            const int m = j + half * 8;
            const float wv = acc[j] / (sRS[w][m] + 1e-8f);   // fold softmax denom
            const float hv = h[(size_t)(m0 + m) * D_LAT + d];
            const float diff = hv - wv;
            cl = fmaf(diff, diff, cl);
        }
    }

    // ---- block reduction and scaled atomic accumulate
    for (int off = 16; off > 0; off >>= 1) cl += __shfl_down(cl, off, 32);
    if (ln == 0) sRed[w] = cl;
    __syncthreads();
    if (tid == 0) {
        float s = 0.0f;
        for (int k = 0; k < WAVES; ++k) s += sRed[k];
        atomicAdd(out, s * scale);
    }
}

extern "C" void kernel_launch(void* const* d_in, const int* in_sizes, int n_in,
                              void* d_out, int out_size, void* d_ws, size_t ws_size,
                              hipStream_t stream) {
    (void)n_in; (void)out_size; (void)d_ws; (void)ws_size;
    const float* x  = (const float*)d_in[0];   // [N, 1024]
    const float* h  = (const float*)d_in[1];   // [N, 128]
    const float* ax = (const float*)d_in[2];   // [N, 1024]
    const float* R  = (const float*)d_in[3];   // [256, 128]
    float* out = (float*)d_out;

    const int n1 = in_sizes[0];                // N * 1024
    const int N  = in_sizes[1] / D_LAT;        // rows

    zero_kernel<<<1, 1, 0, stream>>>(out);

    const int n4 = n1 >> 2;
    recon_kernel<<<2048, 256, 0, stream>>>((const float4*)x, (const float4*)ax,
                                           out, n4, 1.0f / (float)n1);

    const int blocks = N / (WAVES * 16);       // 128 rows per block
    cluster_kernel<<<blocks, 256, 0, stream>>>(h, R, out,
                                               1.0f / ((float)N * (float)D_LAT));
}